// Recurrence_343597384083
// MI455X (gfx1250) — compile-verified
//
#include <hip/hip_runtime.h>
#include <stdint.h>

// Problem constants (from reference: B=4, S=8192, D=1024, fp32)
#define B_ 4
#define S_ 8192
#define D_ 1024
#define C_ 128              // number of time chunks
#define CH_ (S_ / C_)       // 64 timesteps per chunk
#define TT_ 4               // timesteps per LDS tile
#define NT_ (CH_ / TT_)     // 16 tiles per chunk
#define THREADS_ 256        // 4 channels per thread -> covers D=1024

// ---------------------------------------------------------------------------
// CDNA5 async global->LDS copy (gfx1250), GVS addressing form:
//   mem_addr = SGPR_base + VGPR_offset32 ; LDS dest addr in VGPR.
// Keeps the uniform chunk base in SGPRs -> no per-lane 64-bit address math.
// Completion tracked by ASYNCcnt (s_wait_asynccnt).
// ---------------------------------------------------------------------------
__device__ __forceinline__ void async_ld_b128(uint32_t lds_byte_addr,
                                              uint32_t voff_bytes,
                                              const float* sbase) {
  asm volatile("global_load_async_to_lds_b128 %0, %1, %2"
               :: "v"(lds_byte_addr), "v"(voff_bytes), "s"(sbase)
               : "memory");
}
#define WAIT_ASYNC_LE(n) asm volatile("s_wait_asynccnt " #n ::: "memory")

// LDS layout: [buf(2)][arr(2: x,alpha)][TT_][D_] floats = 64 KB total
__device__ __forceinline__ uint32_t lds_off(float* smem, int buf, int arr,
                                            int t, int ch) {
  float* p = &smem[(((buf * 2 + arr) * TT_) + t) * D_ + ch];
  return (uint32_t)(uintptr_t)p;   // low 32 bits of generic addr = LDS offset
}

// Issue one tile's async loads: each wave loads ONLY its own lanes' channels,
// so no workgroup barrier is needed, only s_wait_asynccnt.
// 2*TT_ = 8 async instructions per wave per tile (512 B per instr per wave).
__device__ __forceinline__ void issue_tile(float* smem, const float* xg,
                                           const float* ag, int buf, int t0,
                                           int ch0) {
#pragma unroll
  for (int t = 0; t < TT_; ++t) {
    const uint32_t voff = (uint32_t)(((t0 + t) * D_ + ch0) * 4);
    async_ld_b128(lds_off(smem, buf, 0, t, ch0), voff, xg);
    async_ld_b128(lds_off(smem, buf, 1, t, ch0), voff, ag);
  }
}

// ---------------------------------------------------------------------------
// Pass 1: per-(b,chunk) affine aggregate  (A = prod alpha, X = scan from 0)
// ws layout: A at [0,N), X at [N,2N), Hin at [2N,3N), N = B*C*D floats
// ---------------------------------------------------------------------------
__global__ __launch_bounds__(THREADS_) void pass1_aggregate(
    const float* __restrict__ x, const float* __restrict__ alpha,
    float* __restrict__ wsA, float* __restrict__ wsX) {
  extern __shared__ float smem[];
  const int blk = blockIdx.x;
  const int b = blk / C_;
  const int c = blk % C_;
  const int ch0 = (int)threadIdx.x * 4;

  const float* xg = x + ((size_t)b * S_ + (size_t)c * CH_) * D_;
  const float* ag = alpha + ((size_t)b * S_ + (size_t)c * CH_) * D_;

  float h0 = 0.f, h1 = 0.f, h2 = 0.f, h3 = 0.f;
  float A0 = 1.f, A1 = 1.f, A2 = 1.f, A3 = 1.f;

  issue_tile(smem, xg, ag, 0, 0, ch0);
  for (int tile = 0; tile < NT_; ++tile) {
    if (tile + 1 < NT_) {
      issue_tile(smem, xg, ag, (tile + 1) & 1, (tile + 1) * TT_, ch0);
      WAIT_ASYNC_LE(8);   // previous tile's 8 loads done; next 8 in flight
    } else {
      WAIT_ASYNC_LE(0);
    }
    const int buf = tile & 1;
#pragma unroll
    for (int t = 0; t < TT_; ++t) {
      const float4 xv =
          *(const float4*)&smem[((buf * 2 + 0) * TT_ + t) * D_ + ch0];
      const float4 av =
          *(const float4*)&smem[((buf * 2 + 1) * TT_ + t) * D_ + ch0];
      h0 = fmaf(av.x, h0, xv.x); A0 *= av.x;
      h1 = fmaf(av.y, h1, xv.y); A1 *= av.y;
      h2 = fmaf(av.z, h2, xv.z); A2 *= av.z;
      h3 = fmaf(av.w, h3, xv.w); A3 *= av.w;
    }
  }

  const size_t base = ((size_t)b * C_ + c) * D_ + ch0;
  *(float4*)&wsA[base] = make_float4(A0, A1, A2, A3);
  *(float4*)&wsX[base] = make_float4(h0, h1, h2, h3);
}

// ---------------------------------------------------------------------------
// Pass 2: sequential scan over the C chunk aggregates per channel.
// 4096 independent channels, one thread each; ~6 MB of L2-resident traffic.
// Separate __restrict__ regions let the compiler hoist loads over the Hin
// stores and software-pipeline the whole loop (only the FMA chain is serial).
// ---------------------------------------------------------------------------
__global__ __launch_bounds__(THREADS_) void pass2_scan(
    const float* __restrict__ wsA, const float* __restrict__ wsX,
    float* __restrict__ wsH) {
  const int gid = (int)(blockIdx.x * THREADS_ + threadIdx.x);  // 0..B*D-1
  const int b = gid / D_;
  const int d = gid % D_;
  const size_t base0 = (size_t)b * C_ * D_ + d;

  const float* A = wsA + base0;
  const float* X = wsX + base0;
  float* H = wsH + base0;

  float h = 0.f;
#pragma unroll 8
  for (int c = 0; c < C_; ++c) {
    const float a = A[(size_t)c * D_];
    const float xv = X[(size_t)c * D_];
    H[(size_t)c * D_] = h;        // carry-in for chunk c
    h = fmaf(a, h, xv);
  }
}

// ---------------------------------------------------------------------------
// Pass 3: re-stream x/alpha, run recurrence seeded with Hin, write outputs.
// ---------------------------------------------------------------------------
__global__ __launch_bounds__(THREADS_) void pass3_apply(
    const float* __restrict__ x, const float* __restrict__ alpha,
    const float* __restrict__ wsH, float* __restrict__ out) {
  extern __shared__ float smem[];
  const int blk = blockIdx.x;
  const int b = blk / C_;
  const int c = blk % C_;
  const int ch0 = (int)threadIdx.x * 4;

  const float* xg = x + ((size_t)b * S_ + (size_t)c * CH_) * D_;
  const float* ag = alpha + ((size_t)b * S_ + (size_t)c * CH_) * D_;
  float* og = out + ((size_t)b * S_ + (size_t)c * CH_) * D_;

  const size_t base = ((size_t)b * C_ + c) * D_ + ch0;
  const float4 hv = *(const float4*)&wsH[base];
  float h0 = hv.x, h1 = hv.y, h2 = hv.z, h3 = hv.w;

  issue_tile(smem, xg, ag, 0, 0, ch0);
  for (int tile = 0; tile < NT_; ++tile) {
    if (tile + 1 < NT_) {
      issue_tile(smem, xg, ag, (tile + 1) & 1, (tile + 1) * TT_, ch0);
      WAIT_ASYNC_LE(8);
    } else {
      WAIT_ASYNC_LE(0);
    }
    const int buf = tile & 1;
    const int t0 = tile * TT_;
#pragma unroll
    for (int t = 0; t < TT_; ++t) {
      const float4 xv =
          *(const float4*)&smem[((buf * 2 + 0) * TT_ + t) * D_ + ch0];
      const float4 av =
          *(const float4*)&smem[((buf * 2 + 1) * TT_ + t) * D_ + ch0];
      h0 = fmaf(av.x, h0, xv.x);
      h1 = fmaf(av.y, h1, xv.y);
      h2 = fmaf(av.z, h2, xv.z);
      h3 = fmaf(av.w, h3, xv.w);
      *(float4*)&og[(size_t)(t0 + t) * D_ + ch0] =
          make_float4(h0, h1, h2, h3);
    }
  }
}

// ---------------------------------------------------------------------------
extern "C" void kernel_launch(void* const* d_in, const int* in_sizes, int n_in,
                              void* d_out, int out_size, void* d_ws,
                              size_t ws_size, hipStream_t stream) {
  const float* x = (const float*)d_in[0];
  const float* alpha = (const float*)d_in[1];
  float* out = (float*)d_out;
  float* ws = (float*)d_ws;   // needs 3 * B*C*D * 4 bytes = 6 MB

  const size_t N = (size_t)B_ * C_ * D_;
  float* wsA = ws;
  float* wsX = ws + N;
  float* wsH = ws + 2 * N;

  const size_t smem_bytes = (size_t)2 * 2 * TT_ * D_ * sizeof(float);  // 64 KB

  pass1_aggregate<<<B_ * C_, THREADS_, smem_bytes, stream>>>(x, alpha, wsA,
                                                             wsX);
  pass2_scan<<<(B_ * D_) / THREADS_, THREADS_, 0, stream>>>(wsA, wsX, wsH);
  pass3_apply<<<B_ * C_, THREADS_, smem_bytes, stream>>>(x, alpha, wsH, out);
}